// Capsule_5007931867799
// MI455X (gfx1250) — compile-verified
//
#include <hip/hip_runtime.h>
#include <math.h>

// ---------------------------------------------------------------------------
// Capsule routing for MI455X (gfx1250, wave32).
//   u_vecs: [32, 4096, 128] f32   W: [128, 256] f32   out: [32, 16, 16] f32
// GEMM (M=131072, K=128, N=256) via V_WMMA_F32_16X16X4_F32.
//   - W staged into LDS with async global->LDS (global_load_async_to_lds_b128,
//     ASYNCcnt), the CDNA5-native staging path.
//   - B fragments for two 16-col tiles batch-preloaded to registers, then two
//     independent WMMA accumulator chains -> amortized s_wait_dscnt, 2-way ILP.
// Routing: 16-lane-shuffle softmax, deterministic chunked partials (no float
// atomics). Whole problem is HBM-bound (~0.7 GB traffic).
// Workspace: 131072*256*4 B (u_hat) + 65536*4 (partials) + 8192*4 (o).
// ---------------------------------------------------------------------------

typedef __attribute__((ext_vector_type(2))) float v2f;
typedef __attribute__((ext_vector_type(4))) float v4f;
typedef __attribute__((ext_vector_type(8))) float v8f;

#define BATCH 32
#define ISZ   4096
#define KSZ   128
#define COLS  256      // 16 capsules * 16 dims
#define NROWS ((long)BATCH * ISZ)   // 131072

// ---------------------------------------------------------------------------
// zero o_in (8192 floats) so iteration-0 logits are exactly 0 (c = 1/16)
// ---------------------------------------------------------------------------
__global__ void zero_oin(float* __restrict__ o_in) {
    int t = blockIdx.x * blockDim.x + threadIdx.x;
    if (t < BATCH * COLS) o_in[t] = 0.0f;
}

// ---------------------------------------------------------------------------
// GEMM: u_hat[row, col] = sum_d U[row, d] * W[d, col]
// 256 threads = 8 waves. Wave w owns row strip (blockIdx.x*8 + w)*16.
// A fragments (16x128 f32) live in 64 VGPRs/lane; B from LDS copy of W.
// dynamic LDS = 128*256*4 = 128 KB (within the 320 KB WGP LDS).
// ---------------------------------------------------------------------------
__global__ __launch_bounds__(256) void gemm_uhat(const float* __restrict__ U,
                                                 const float* __restrict__ W,
                                                 float* __restrict__ uhat) {
    extern __shared__ float Wlds[];  // [128][256], dynamic LDS (byte offset 0)

    const int tid  = threadIdx.x;
    const int wave = tid >> 5;
    const int lane = tid & 31;
    const int l16  = lane & 15;
    const int hi   = lane >> 4;          // which K-pair half of the A fragment

    const long strip = (long)blockIdx.x * 8 + wave;
    const long row0  = strip * 16;

    // Async-stage W (128 KB) into LDS: 256 threads * 32 * b128 = 131072 B.
    // Kernel has no static LDS, so dynamic LDS starts at byte offset 0 and the
    // per-lane LDS destination VGPR is just the byte offset.
    {
        unsigned long g = (unsigned long)W + (unsigned)tid * 16u;
        unsigned      l = (unsigned)tid * 16u;
#pragma unroll
        for (int j = 0; j < 32; ++j) {
            unsigned long gj = g + (unsigned)j * 4096u;
            unsigned      lj = l + (unsigned)j * 4096u;
            asm volatile("global_load_async_to_lds_b128 %0, %1, off"
                         :: "v"(lj), "v"(gj) : "memory");
        }
    }

    // Issue A loads while the async DMA runs.
    // A layout for V_WMMA_F32_16X16X4_F32: lanes 0-15 hold K={0,1},
    // lanes 16-31 hold K={2,3}; M = lane%16.
    const float* Arow = U + (row0 + l16) * (long)KSZ;
    v2f a[32];
#pragma unroll
    for (int kk = 0; kk < 32; ++kk)
        a[kk] = *(const v2f*)(Arow + 4 * kk + 2 * hi);

    asm volatile("s_wait_asynccnt 0" ::: "memory");  // this wave's DMA done
    __syncthreads();                                  // all waves' slices done

    // Two column tiles per pass: batch all B fragments into registers first,
    // then run two independent WMMA accumulation chains (2-way ILP).
    for (int nt = 0; nt < 16; nt += 2) {
        v2f b0[32], b1[32];
#pragma unroll
        for (int kk = 0; kk < 32; ++kk) {
            // B layout: VGPR0 = rows {0,2}, VGPR1 = rows {1,3}; N = lane%16.
            const int r0 = 4 * kk + 2 * hi;
            b0[kk].x = Wlds[r0 * COLS + nt * 16 + l16];
            b0[kk].y = Wlds[(r0 + 1) * COLS + nt * 16 + l16];
            b1[kk].x = Wlds[r0 * COLS + (nt + 1) * 16 + l16];
            b1[kk].y = Wlds[(r0 + 1) * COLS + (nt + 1) * 16 + l16];
        }
        v8f acc0 = {};
        v8f acc1 = {};
#pragma unroll
        for (int kk = 0; kk < 32; ++kk) {
            acc0 = __builtin_amdgcn_wmma_f32_16x16x4_f32(
                false, a[kk], false, b0[kk], (short)0, acc0, false, false);
            acc1 = __builtin_amdgcn_wmma_f32_16x16x4_f32(
                false, a[kk], false, b1[kk], (short)0, acc1, false, false);
        }
        // C layout: VGPR j holds rows {j, j+8}; N = lane%16.
        float* out0 = uhat + (row0 + hi * 8) * COLS + nt * 16 + l16;
        float* out1 = out0 + 16;
#pragma unroll
        for (int j = 0; j < 8; ++j) {
            out0[(long)j * COLS] = acc0[j];
            out1[(long)j * COLS] = acc1[j];
        }
    }
}

// ---------------------------------------------------------------------------
// One routing iteration. grid = (32 batches, 8 chunks), 256 threads (8 waves).
// Lane n (= lane&15) owns capsule n; half-waves process two i's per step.
// Writes deterministic per-(b,chunk) partials: part[(b*8+chunk)*256 + n*16+d].
// ---------------------------------------------------------------------------
__global__ __launch_bounds__(256) void routing(const float* __restrict__ uhat,
                                               const float* __restrict__ o_in,
                                               float* __restrict__ part) {
    __shared__ float red[8 * COLS];

    const int tid  = threadIdx.x;
    const int wave = tid >> 5;
    const int lane = tid & 31;
    const int n    = lane & 15;
    const int hi   = lane >> 4;
    const int b    = blockIdx.x;
    const int chunk = blockIdx.y;

    // o vector for (b, n): 16 floats, kept in registers for the whole chunk.
    float o[16];
    {
        const float* op = o_in + b * COLS + n * 16;
#pragma unroll
        for (int d = 0; d < 16; ++d) o[d] = op[d];
    }

    float acc[16];
#pragma unroll
    for (int d = 0; d < 16; ++d) acc[d] = 0.0f;

    const float* base = uhat + ((long)b * ISZ) * COLS;
    const int i0 = chunk * 512 + wave * 64 + hi;   // 64 i's per wave, 2 per step

    for (int step = 0; step < 32; ++step) {
        const int i = i0 + 2 * step;
        const float* uh = base + (long)i * COLS + n * 16;
        v4f u0 = *(const v4f*)(uh + 0);
        v4f u1 = *(const v4f*)(uh + 4);
        v4f u2 = *(const v4f*)(uh + 8);
        v4f u3 = *(const v4f*)(uh + 12);

        // logit[n] = dot(o[n,:], u_hat[b,i,n,:])
        float lg = u0.x * o[0]  + u0.y * o[1]  + u0.z * o[2]  + u0.w * o[3]
                 + u1.x * o[4]  + u1.y * o[5]  + u1.z * o[6]  + u1.w * o[7]
                 + u2.x * o[8]  + u2.y * o[9]  + u2.z * o[10] + u2.w * o[11]
                 + u3.x * o[12] + u3.y * o[13] + u3.z * o[14] + u3.w * o[15];

        // softmax over the 16 capsules (16-lane groups, wave32 shuffles)
        float m = lg;
#pragma unroll
        for (int mk = 8; mk; mk >>= 1) m = fmaxf(m, __shfl_xor(m, mk, 16));
        float e = __expf(lg - m);
        float s = e;
#pragma unroll
        for (int mk = 8; mk; mk >>= 1) s += __shfl_xor(s, mk, 16);
        const float c = e / s;

        acc[0]  += c * u0.x;  acc[1]  += c * u0.y;  acc[2]  += c * u0.z;  acc[3]  += c * u0.w;
        acc[4]  += c * u1.x;  acc[5]  += c * u1.y;  acc[6]  += c * u1.z;  acc[7]  += c * u1.w;
        acc[8]  += c * u2.x;  acc[9]  += c * u2.y;  acc[10] += c * u2.z;  acc[11] += c * u2.w;
        acc[12] += c * u3.x;  acc[13] += c * u3.y;  acc[14] += c * u3.z;  acc[15] += c * u3.w;
    }

    // fold the two i-halves (lane n <-> lane n+16)
#pragma unroll
    for (int d = 0; d < 16; ++d) acc[d] += __shfl_xor(acc[d], 16, 32);

    if (hi == 0) {
        float* r = red + wave * COLS + n * 16;
#pragma unroll
        for (int d = 0; d < 16; ++d) r[d] = acc[d];
    }
    __syncthreads();

    // tid = n*16+d; fixed-order sum over 8 waves -> deterministic partial
    float s = 0.0f;
#pragma unroll
    for (int w = 0; w < 8; ++w) s += red[w * COLS + tid];
    part[(b * 8 + chunk) * COLS + tid] = s;
}

// ---------------------------------------------------------------------------
// Sum the 8 chunk partials (fixed order), then l2_normalize (iters 0..2) or
// squash (last iter). grid = 32 blocks, 256 threads; tid = n*16+d.
// ---------------------------------------------------------------------------
__global__ __launch_bounds__(256) void finalize(const float* __restrict__ part,
                                                float* __restrict__ o_next,
                                                float* __restrict__ out,
                                                int last) {
    const int b  = blockIdx.x;
    const int nd = threadIdx.x;

    float v = 0.0f;
#pragma unroll
    for (int c = 0; c < 8; ++c) v += part[(b * 8 + c) * COLS + nd];

    float sq = v * v;   // sum of squares over the 16 dims of this capsule
#pragma unroll
    for (int mk = 8; mk; mk >>= 1) sq += __shfl_xor(sq, mk, 16);

    if (!last) {
        o_next[b * COLS + nd] = v * rsqrtf(fmaxf(sq, 1e-12f));
    } else {
        const float ss = sq + 1e-7f;
        out[b * COLS + nd] = v * (sqrtf(ss) / (0.5f + ss));
    }
}

// ---------------------------------------------------------------------------
extern "C" void kernel_launch(void* const* d_in, const int* in_sizes, int n_in,
                              void* d_out, int out_size, void* d_ws, size_t ws_size,
                              hipStream_t stream) {
    const float* U = (const float*)d_in[0];   // [32, 4096, 128]
    const float* W = (const float*)d_in[1];   // [128, 256]
    float* out = (float*)d_out;               // [32, 16, 16]

    float* uhat = (float*)d_ws;                       // 131072*256 floats (128 MB)
    float* part = uhat + NROWS * COLS;                // 256*256 floats
    float* o_in = part + (size_t)BATCH * 8 * COLS;    // 8192 floats

    zero_oin<<<32, 256, 0, stream>>>(o_in);

    // 8192 row strips / 8 waves per block -> 1024 blocks; 128 KB dynamic LDS
    gemm_uhat<<<1024, 256, KSZ * COLS * sizeof(float), stream>>>(U, W, uhat);

    for (int it = 0; it < 4; ++it) {
        routing<<<dim3(BATCH, 8), 256, 0, stream>>>(uhat, o_in, part);
        finalize<<<BATCH, 256, 0, stream>>>(part, o_in, out, it == 3 ? 1 : 0);
    }
}